// GraphConvEncoder_89635967467603
// MI455X (gfx1250) — compile-verified
//
#include <hip/hip_runtime.h>
#include <hip/hip_bf16.h>

// ---- problem constants -----------------------------------------------------
#define NODES   100000
#define FEAT_IN 128
#define FEAT_H  128
#define FEAT_O  64

typedef float v2f __attribute__((ext_vector_type(2)));
typedef float v8f __attribute__((ext_vector_type(8)));

// ---- zero a float buffer ---------------------------------------------------
__global__ void zero_kernel(float4* __restrict__ p, long long n4) {
    long long i = (long long)blockIdx.x * blockDim.x + threadIdx.x;
    long long stride = (long long)gridDim.x * blockDim.x;
    float4 z = {0.f, 0.f, 0.f, 0.f};
    for (; i < n4; i += stride) p[i] = z;
}

// ---- edge scatter-add: agg[dst] += feat[src] -------------------------------
// One wave (32 lanes) per edge; lane l handles features [4l .. 4l+3] of the
// 128-wide row -> coalesced 512B gather per edge, f32 atomics land in L2
// (feature matrix is 51MB, fully L2-resident on MI455X's 192MB L2).
__global__ void scatter_kernel(const float* __restrict__ feat,
                               const int* __restrict__ srcIdx,
                               const int* __restrict__ dstIdx,
                               float* __restrict__ agg,
                               int E) {
    int wavesPerBlock = blockDim.x >> 5;
    int wave = blockIdx.x * wavesPerBlock + (threadIdx.x >> 5);
    int lane = threadIdx.x & 31;
    if (wave >= E) return;
    int s = srcIdx[wave];
    int d = dstIdx[wave];
    const float4 v = *(const float4*)(feat + (size_t)s * FEAT_IN + lane * 4);
    float* p = agg + (size_t)d * FEAT_IN + lane * 4;
    __hip_atomic_fetch_add(p + 0, v.x, __ATOMIC_RELAXED, __HIP_MEMORY_SCOPE_AGENT);
    __hip_atomic_fetch_add(p + 1, v.y, __ATOMIC_RELAXED, __HIP_MEMORY_SCOPE_AGENT);
    __hip_atomic_fetch_add(p + 2, v.z, __ATOMIC_RELAXED, __HIP_MEMORY_SCOPE_AGENT);
    __hip_atomic_fetch_add(p + 3, v.w, __ATOMIC_RELAXED, __HIP_MEMORY_SCOPE_AGENT);
}

// ---- dense: out = act(agg @ Wrel^T + b + xin @ Wroot^T) --------------------
// One wave computes one 16x16 output tile via V_WMMA_F32_16X16X4_F32,
// stepping K=128 in chunks of 4. Both products share one f32 accumulator.
//
// f32 16x16x4 fragment layout (wave32):
//   A (16x4, M-major): lanes 0-15 -> row M=lane, K={k,k+1} as float2
//                      lanes 16-31 -> row M=lane-16, K={k+2,k+3}
//   B (4x16): B[k][n] = W[n][k] -> same pattern with N in place of M
//   C/D: VGPR r, lanes 0-15 -> (M=r, N=lane); lanes 16-31 -> (M=r+8, N=lane-16)
template <int K, int WOUT, bool RELU>
__global__ void dense_kernel(const float* __restrict__ agg,
                             const float* __restrict__ xin,
                             const float* __restrict__ Wrel,
                             const float* __restrict__ bias,
                             const float* __restrict__ Wroot,
                             float* __restrict__ out,
                             int nTiles) {
    constexpr int tilesF = WOUT / 16;
    int wavesPerBlock = blockDim.x >> 5;
    int wave = blockIdx.x * wavesPerBlock + (threadIdx.x >> 5);
    if (wave >= nTiles) return;          // uniform per-wave: EXEC stays all-1s
    int lane = threadIdx.x & 31;
    int tM = wave / tilesF;
    int tN = wave - tM * tilesF;
    int m0 = tM * 16;
    int n0 = tN * 16;
    int half = lane >> 4;                // 0: lanes 0-15, 1: lanes 16-31
    int l = lane & 15;
    const float* arow = agg  + (size_t)(m0 + l) * K + 2 * half; // A rel row
    const float* xrow = xin  + (size_t)(m0 + l) * K + 2 * half; // A root row
    const float* wrel = Wrel  + (size_t)(n0 + l) * K + 2 * half; // B rel col
    const float* wrt  = Wroot + (size_t)(n0 + l) * K + 2 * half; // B root col

    v8f acc = {};
#pragma unroll
    for (int k = 0; k < K; k += 4) {
        v2f a0 = *(const v2f*)(arow + k);
        v2f b0 = *(const v2f*)(wrel + k);
        acc = __builtin_amdgcn_wmma_f32_16x16x4_f32(false, a0, false, b0,
                                                    (short)0, acc, false, false);
        v2f a1 = *(const v2f*)(xrow + k);
        v2f b1 = *(const v2f*)(wrt + k);
        acc = __builtin_amdgcn_wmma_f32_16x16x4_f32(false, a1, false, b1,
                                                    (short)0, acc, false, false);
    }

    float bv = bias[n0 + l];
#pragma unroll
    for (int r = 0; r < 8; ++r) {
        float v = acc[r] + bv;
        if (RELU) v = fmaxf(v, 0.f);
        out[(size_t)(m0 + r + half * 8) * WOUT + n0 + l] = v;
    }
}

extern "C" void kernel_launch(void* const* d_in, const int* in_sizes, int n_in,
                              void* d_out, int out_size, void* d_ws, size_t ws_size,
                              hipStream_t stream) {
    const float* x      = (const float*)d_in[0];
    const int*   eidx   = (const int*)d_in[1];
    const float* W1_rel = (const float*)d_in[2];
    const float* b1     = (const float*)d_in[3];
    const float* W1_rt  = (const float*)d_in[4];
    const float* W2_rel = (const float*)d_in[5];
    const float* b2     = (const float*)d_in[6];
    const float* W2_rt  = (const float*)d_in[7];
    float* out = (float*)d_out;

    const int E = in_sizes[1] / 2;
    const int* srcI = eidx;
    const int* dstI = eidx + E;

    // workspace: agg [N x 128] then h [N x 128]
    float* agg = (float*)d_ws;
    float* h   = agg + (size_t)NODES * FEAT_IN;

    const long long aggN4 = (long long)NODES * FEAT_IN / 4;
    const int ZB = 256;
    int zgrid = 4096;

    const int waveBlk = 256;              // 8 waves/block (wave32)
    int scatterBlocks = (E + 7) / 8;

    // ---- layer 1 ----
    zero_kernel<<<zgrid, ZB, 0, stream>>>((float4*)agg, aggN4);
    scatter_kernel<<<scatterBlocks, waveBlk, 0, stream>>>(x, srcI, dstI, agg, E);
    {
        int tiles = (NODES / 16) * (FEAT_H / 16);      // 6250 * 8 = 50000
        int blocks = (tiles + 7) / 8;
        dense_kernel<FEAT_IN, FEAT_H, true>
            <<<blocks, waveBlk, 0, stream>>>(agg, x, W1_rel, b1, W1_rt, h, tiles);
    }

    // ---- layer 2 ----
    zero_kernel<<<zgrid, ZB, 0, stream>>>((float4*)agg, aggN4);
    scatter_kernel<<<scatterBlocks, waveBlk, 0, stream>>>(h, srcI, dstI, agg, E);
    {
        int tiles = (NODES / 16) * (FEAT_O / 16);      // 6250 * 4 = 25000
        int blocks = (tiles + 7) / 8;
        dense_kernel<FEAT_H, FEAT_O, false>
            <<<blocks, waveBlk, 0, stream>>>(agg, h, W2_rel, b2, W2_rt, out, tiles);
    }
}